// Prediction_57887569215535
// MI455X (gfx1250) — compile-verified
//
#include <hip/hip_runtime.h>
#include <float.h>

#define HM_B 16
#define HM_C 80
#define HM_H 256
#define HM_W 256
#define HW_   (HM_H * HM_W)       // 65536
#define TOPK 100
#define THRESH 0.01f
#define SCALE 4.0f

#define STRIP 32
#define LROWS (STRIP + 2)         // 34 rows staged (with top/bottom halo)
#define PT 8                      // per-thread top-8
#define PLANE_K 16                // per-plane top-16
#define NTHREADS 256
#define NWAVES (NTHREADS / 32)

#if defined(__has_builtin)
#if __has_builtin(__builtin_amdgcn_global_load_async_to_lds_b128)
#define HAS_ASYNC 1
#endif
#endif

typedef int v4i_t __attribute__((ext_vector_type(4)));
#ifdef HAS_ASYNC
typedef __attribute__((address_space(1))) v4i_t* gptr_v4i;
typedef __attribute__((address_space(3))) v4i_t* lptr_v4i;
#endif

__device__ __forceinline__ bool better(float s0, unsigned i0, float s1, unsigned i1) {
  // descending score, ascending flat index on ties (matches jax.lax.top_k order)
  return (s0 > s1) || (s0 == s1 && i0 < i1);
}

__device__ __forceinline__ void wait_async_then_barrier() {
#ifdef HAS_ASYNC
#if __has_builtin(__builtin_amdgcn_s_wait_asynccnt)
  __builtin_amdgcn_s_wait_asynccnt(0);
#else
  asm volatile("s_wait_asynccnt 0" ::: "memory");
#endif
#endif
  __syncthreads();
}

// ---------------- Kernel 1: per-plane NMS + top-16 ----------------
extern "C" __global__ __launch_bounds__(NTHREADS)
void k_plane_topk(const float* __restrict__ hm,
                  float* __restrict__ wsS, unsigned* __restrict__ wsI) {
  __shared__ __align__(16) float tile[LROWS * HM_W];   // 34 KB strip
  __shared__ float    cs[NTHREADS * PT];               // 8 KB candidates
  __shared__ unsigned ci[NTHREADS * PT];               // 8 KB
  __shared__ float    redS[NWAVES];
  __shared__ unsigned redI[NWAVES];
  __shared__ unsigned redL[NWAVES];
  __shared__ float    outS[PLANE_K];
  __shared__ unsigned outI[PLANE_K];

  const int plane = blockIdx.x;           // plane = b*80 + c  (heatmap is B,C,H,W)
  const int tid   = threadIdx.x;
  const float* pbase = hm + (size_t)plane * HW_;

  float    bs[PT];
  unsigned bi[PT];
#pragma unroll
  for (int k = 0; k < PT; ++k) { bs[k] = -1e30f; bi[k] = 0xFFFFFFFFu; }

  const int  t   = tid;                    // thread owns column t
  const bool lok = (t > 0), rok = (t < HM_W - 1);

  for (int s0 = 0; s0 < HM_H; s0 += STRIP) {
    __syncthreads();   // previous strip fully consumed before overwrite
    // stage rows [s0-1, s0+STRIP] into LDS; out-of-plane rows -> -inf fill
    for (int chunk = tid; chunk < LROWS * (HM_W / 4); chunk += NTHREADS) {
      const int r  = chunk >> 6;           // 64 float4 chunks per 256-wide row
      const int c4 = chunk & 63;
      const int pr = s0 - 1 + r;
      const int le = r * HM_W + c4 * 4;
      if (pr >= 0 && pr < HM_H) {
#ifdef HAS_ASYNC
        float* gsrc = const_cast<float*>(pbase + (size_t)pr * HM_W + c4 * 4);
        __builtin_amdgcn_global_load_async_to_lds_b128(
            (gptr_v4i)gsrc, (lptr_v4i)(&tile[le]), 0, 0);
#else
        *(float4*)(&tile[le]) = *(const float4*)(pbase + (size_t)pr * HM_W + c4 * 4);
#endif
      } else {
        float4 f; f.x = f.y = f.z = f.w = -1e30f;
        *(float4*)(&tile[le]) = f;
      }
    }
    wait_async_then_barrier();

    for (int sr = 0; sr < STRIP; ++sr) {
      const int bT = sr * HM_W + t, bM = bT + HM_W, bB = bM + HM_W;
      const float c0 = tile[bM];
      float m = fmaxf(c0, fmaxf(tile[bT], tile[bB]));
      if (lok) m = fmaxf(m, fmaxf(tile[bT - 1], fmaxf(tile[bM - 1], tile[bB - 1])));
      if (rok) m = fmaxf(m, fmaxf(tile[bT + 1], fmaxf(tile[bM + 1], tile[bB + 1])));
      if (c0 == m && c0 > THRESH) {        // peak; sub-threshold ones are fill-equivalent
        float s = c0;
        unsigned i = (unsigned)((s0 + sr) * HM_W + t);
        if (better(s, i, bs[PT - 1], bi[PT - 1])) {
#pragma unroll
          for (int k = 0; k < PT; ++k) {
            if (better(s, i, bs[k], bi[k])) {
              float ts = bs[k]; unsigned ti = bi[k];
              bs[k] = s; bi[k] = i; s = ts; i = ti;
            }
          }
        }
      }
    }
  }

#pragma unroll
  for (int k = 0; k < PT; ++k) { cs[tid * PT + k] = bs[k]; ci[tid * PT + k] = bi[k]; }
  __syncthreads();

  // extract plane top-16 (exact over the 2048 pooled candidates)
  for (int it = 0; it < PLANE_K; ++it) {
    float s = -3e30f; unsigned i = 0xFFFFFFFFu; unsigned l = 0;
#pragma unroll
    for (int k = 0; k < PT; ++k) {
      const int e = tid * PT + k;
      const float sv = cs[e]; const unsigned iv = ci[e];
      if (better(sv, iv, s, i)) { s = sv; i = iv; l = (unsigned)e; }
    }
#pragma unroll
    for (int off = 16; off > 0; off >>= 1) {
      float    os = __shfl_xor(s, off, 32);
      unsigned oi = __shfl_xor(i, off, 32);
      unsigned ol = __shfl_xor(l, off, 32);
      if (better(os, oi, s, i)) { s = os; i = oi; l = ol; }
    }
    if ((tid & 31) == 0) { const int w = tid >> 5; redS[w] = s; redI[w] = i; redL[w] = l; }
    __syncthreads();
    if (tid == 0) {
      float bsc = redS[0]; unsigned bix = redI[0], blt = redL[0];
#pragma unroll
      for (int w = 1; w < NWAVES; ++w)
        if (better(redS[w], redI[w], bsc, bix)) { bsc = redS[w]; bix = redI[w]; blt = redL[w]; }
      outS[it] = bsc; outI[it] = bix;
      cs[blt] = -3e30f;                    // retire winner
    }
    __syncthreads();
  }

  if (tid < PLANE_K) {
    const unsigned c = (unsigned)(plane % HM_C);
    const unsigned i = outI[tid];
    wsS[plane * PLANE_K + tid] = outS[tid];
    wsI[plane * PLANE_K + tid] = (i == 0xFFFFFFFFu) ? 0xFFFFFFFFu : (c * (unsigned)HW_ + i);
  }
}

// ---------------- Kernel 2: per-batch merge + decode ----------------
#define K2_N  (HM_C * PLANE_K)        // 1280 candidates/batch
#define K2_PER (K2_N / NTHREADS)      // 5 per thread

extern "C" __global__ __launch_bounds__(NTHREADS)
void k_batch_out(const float* __restrict__ wsS, const unsigned* __restrict__ wsI,
                 const float* __restrict__ offset, const float* __restrict__ wh,
                 float* __restrict__ out) {
  __shared__ float    cs[K2_N];
  __shared__ unsigned ci[K2_N];
  __shared__ float    redS[NWAVES];
  __shared__ unsigned redI[NWAVES];
  __shared__ unsigned redL[NWAVES];
  __shared__ float    topS[TOPK];
  __shared__ unsigned topI[TOPK];

  const int b = blockIdx.x, tid = threadIdx.x;
  for (int e = tid; e < K2_N; e += NTHREADS) {
    cs[e] = wsS[b * K2_N + e];
    ci[e] = wsI[b * K2_N + e];
  }
  __syncthreads();

  for (int r = 0; r < TOPK; ++r) {
    float s = -3e30f; unsigned i = 0xFFFFFFFFu; unsigned l = 0;
#pragma unroll
    for (int k = 0; k < K2_PER; ++k) {
      const int e = tid * K2_PER + k;
      const float sv = cs[e]; const unsigned iv = ci[e];
      if (better(sv, iv, s, i)) { s = sv; i = iv; l = (unsigned)e; }
    }
#pragma unroll
    for (int off = 16; off > 0; off >>= 1) {
      float    os = __shfl_xor(s, off, 32);
      unsigned oi = __shfl_xor(i, off, 32);
      unsigned ol = __shfl_xor(l, off, 32);
      if (better(os, oi, s, i)) { s = os; i = oi; l = ol; }
    }
    if ((tid & 31) == 0) { const int w = tid >> 5; redS[w] = s; redI[w] = i; redL[w] = l; }
    __syncthreads();
    if (tid == 0) {
      float bsc = redS[0]; unsigned bix = redI[0], blt = redL[0];
#pragma unroll
      for (int w = 1; w < NWAVES; ++w)
        if (better(redS[w], redI[w], bsc, bix)) { bsc = redS[w]; bix = redI[w]; blt = redL[w]; }
      topS[r] = bsc; topI[r] = bix;
      cs[blt] = -3e30f;
    }
    __syncthreads();
  }

  if (tid < TOPK) {
    const float s = topS[tid];
    const unsigned fi = topI[tid];
    float* ids_o = out;                         // (16,100,1)
    float* sc_o  = out + HM_B * TOPK;           // (16,100,1)
    float* bb_o  = out + 2 * HM_B * TOPK;       // (16,100,4)
    const int o = b * TOPK + tid;
    if (s > THRESH) {
      const unsigned c   = fi >> 16;            // / 65536
      const unsigned pix = fi & 0xFFFFu;
      const unsigned y = pix >> 8, x = pix & 0xFFu;
      const float* offb = offset + (size_t)b * 2 * HW_;
      const float* whb  = wh     + (size_t)b * 2 * HW_;
      const float ox = offb[pix], oy = offb[HW_ + pix];
      const float ww = whb[pix],  hh = whb[HW_ + pix];
      const float cx = (float)x + ox, cy = (float)y + oy;
      const float hw2 = ww * 0.5f, hh2 = hh * 0.5f;
      ids_o[o] = (float)c;
      sc_o[o]  = s;
      bb_o[o * 4 + 0] = (cx - hw2) * SCALE;
      bb_o[o * 4 + 1] = (cy - hh2) * SCALE;
      bb_o[o * 4 + 2] = (cx + hw2) * SCALE;
      bb_o[o * 4 + 3] = (cy + hh2) * SCALE;
    } else {
      ids_o[o] = -1.0f;
      sc_o[o]  = -1.0f;
      bb_o[o * 4 + 0] = -SCALE;
      bb_o[o * 4 + 1] = -SCALE;
      bb_o[o * 4 + 2] = -SCALE;
      bb_o[o * 4 + 3] = -SCALE;
    }
  }
}

extern "C" void kernel_launch(void* const* d_in, const int* in_sizes, int n_in,
                              void* d_out, int out_size, void* d_ws, size_t ws_size,
                              hipStream_t stream) {
  (void)in_sizes; (void)n_in; (void)out_size; (void)ws_size;
  const float* hm     = (const float*)d_in[0];
  const float* offset = (const float*)d_in[1];
  const float* wh     = (const float*)d_in[2];
  float* out = (float*)d_out;

  float*    wsS = (float*)d_ws;
  unsigned* wsI = (unsigned*)((char*)d_ws + (size_t)HM_B * HM_C * PLANE_K * sizeof(float));

  k_plane_topk<<<HM_B * HM_C, NTHREADS, 0, stream>>>(hm, wsS, wsI);
  k_batch_out<<<HM_B, NTHREADS, 0, stream>>>(wsS, wsI, offset, wh, out);
}